// GAT_16037407884011
// MI455X (gfx1250) — compile-verified
//
#include <hip/hip_runtime.h>
#include <hip/hip_bf16.h>

#define N_NODES   50000
#define N_EDGES   800000
#define IN_DIM    256
#define OUT_DIM   64
#define NEG_SLOPE 0.01f
#define M_TILES   (N_NODES / 16)   // 3125 (exact)

typedef float v2f __attribute__((ext_vector_type(2)));
typedef float v8f __attribute__((ext_vector_type(8)));

// ---------------------------------------------------------------------------
// Kernel 1: z = h @ W^T  via V_WMMA_F32_16X16X4_F32 (full f32 precision).
// One wave computes one 16x16 tile of z. Block = 8 waves = 2 M-tiles x 4 N-tiles.
// A (16x4 f32): lane l<16 -> row M=l, VGPR0/1 = K0/K1; lane l+16 -> K2/K3.
// B (4x16 f32): lane l<16 -> col N=l, VGPR0/1 = K0/K1; lane l+16 -> K2/K3.
// D (16x16 f32, 8 VGPRs): VGPR r -> M = r + 8*(lane>=16), N = lane%16.
// ---------------------------------------------------------------------------
__global__ __launch_bounds__(256) void gat_gemm_wmma(const float* __restrict__ h,
                                                     const float* __restrict__ W,
                                                     float* __restrict__ z) {
  const int lane  = threadIdx.x & 31;
  const int wave  = threadIdx.x >> 5;           // 0..7
  const int ntile = wave & 3;                   // 0..3 -> 16-col slice of OUT_DIM
  const int mtile = blockIdx.x * 2 + (wave >> 2);
  if (mtile >= M_TILES) return;                 // wave-uniform exit

  const int arow  = mtile * 16 + (lane & 15);   // node row this lane feeds A with
  const int bcol  = ntile * 16 + (lane & 15);   // output channel (row of W)
  const int khalf = (lane >> 4) * 2;            // lanes 16..31 carry K+2,K+3

  const float* __restrict__ hrow = h + (size_t)arow * IN_DIM + khalf;
  const float* __restrict__ wrow = W + (size_t)bcol * IN_DIM + khalf;

  v8f acc = {};
#pragma unroll 8
  for (int k = 0; k < IN_DIM; k += 4) {
    v2f a = *(const v2f*)(hrow + k);   // global_load_b64
    v2f b = *(const v2f*)(wrow + k);   // W is 64KB -> L2/L0 resident, shared by M-tiles
    acc = __builtin_amdgcn_wmma_f32_16x16x4_f32(
        /*neg_a=*/false, a, /*neg_b=*/false, b,
        /*c_mod=*/(short)0, acc, /*reuse_a=*/false, /*reuse_b=*/false);
  }

  const int mbase = mtile * 16 + (lane >> 4) * 8;
  const int col   = ntile * 16 + (lane & 15);
#pragma unroll
  for (int r = 0; r < 8; ++r)
    z[(size_t)(mbase + r) * OUT_DIM + col] = acc[r];
}

// ---------------------------------------------------------------------------
// Kernel 2: init accumulators (h_out = 0, m_enc = 0 (== -inf encoded), s = 0)
// ---------------------------------------------------------------------------
__global__ void gat_init(float* __restrict__ hout, unsigned* __restrict__ menc,
                         float* __restrict__ s) {
  int i = blockIdx.x * blockDim.x + threadIdx.x;
  if (i < N_NODES * OUT_DIM) hout[i] = 0.0f;
  if (i < N_NODES) { menc[i] = 0u; s[i] = 0.0f; }
}

// ---------------------------------------------------------------------------
// Kernel 3: el[n] = z[n,:]·a_l ; er[n] = z[n,:]·a_r   (one thread per node)
// ---------------------------------------------------------------------------
__global__ void gat_scores(const float* __restrict__ z, const float* __restrict__ a,
                           float* __restrict__ el, float* __restrict__ er) {
  int n = blockIdx.x * blockDim.x + threadIdx.x;
  if (n >= N_NODES) return;
  const float* zr = z + (size_t)n * OUT_DIM;
  float sl = 0.0f, sr = 0.0f;
#pragma unroll 8
  for (int i = 0; i < OUT_DIM; ++i) {
    float v = zr[i];
    sl += v * a[i];
    sr += v * a[OUT_DIM + i];
  }
  el[n] = sl;
  er[n] = sr;
}

// Order-preserving float <-> uint encoding for atomicMax-based segment_max.
__device__ __forceinline__ unsigned fenc(float f) {
  unsigned u = __float_as_uint(f);
  return (u >> 31) ? ~u : (u | 0x80000000u);
}
__device__ __forceinline__ float fdec(unsigned u) {
  return __uint_as_float((u >> 31) ? (u ^ 0x80000000u) : ~u);
}

// ---------------------------------------------------------------------------
// Kernel 4: e = leaky_relu(el[src]+er[dst]); segment_max via u32 atomicMax
// ---------------------------------------------------------------------------
__global__ void gat_edge_max(const int* __restrict__ src, const int* __restrict__ dst,
                             const float* __restrict__ el, const float* __restrict__ er,
                             float* __restrict__ e, unsigned* __restrict__ menc) {
  int i = blockIdx.x * blockDim.x + threadIdx.x;
  if (i >= N_EDGES) return;
  int sn = src[i], d = dst[i];
  float v = el[sn] + er[d];
  v = (v > 0.0f) ? v : v * NEG_SLOPE;
  e[i] = v;
  atomicMax(menc + d, fenc(v));
}

// ---------------------------------------------------------------------------
// Kernel 5: ex = exp(e - m[dst]); segment_sum via f32 atomicAdd
// ---------------------------------------------------------------------------
__global__ void gat_edge_exp(const int* __restrict__ dst, const unsigned* __restrict__ menc,
                             float* __restrict__ e, float* __restrict__ s) {
  int i = blockIdx.x * blockDim.x + threadIdx.x;
  if (i >= N_EDGES) return;
  int d = dst[i];
  float ex = __expf(e[i] - fdec(menc[d]));
  e[i] = ex;
  atomicAdd(s + d, ex);
}

// ---------------------------------------------------------------------------
// Kernel 6: h_out[dst] += (ex/s[dst]) * z[src]. One wave per edge, 2 dims/lane.
// z and h_out are L2-resident (12.8 MB each), so gather + atomics stay on-die.
// ---------------------------------------------------------------------------
__global__ __launch_bounds__(256) void gat_scatter(const int* __restrict__ src,
                                                   const int* __restrict__ dst,
                                                   const float* __restrict__ e,
                                                   const float* __restrict__ s,
                                                   const float* __restrict__ z,
                                                   float* __restrict__ hout) {
  int edge = (blockIdx.x * blockDim.x + threadIdx.x) >> 5;
  int lane = threadIdx.x & 31;
  if (edge >= N_EDGES) return;
  int sn = src[edge], d = dst[edge];
  float alpha = e[edge] / s[d];
  v2f zv = *(const v2f*)(z + (size_t)sn * OUT_DIM + lane * 2);
  float* orow = hout + (size_t)d * OUT_DIM + lane * 2;
  atomicAdd(orow,     alpha * zv.x);
  atomicAdd(orow + 1, alpha * zv.y);
}

// ---------------------------------------------------------------------------
// Launch: inputs = {h, W, a, src, dst}; output = h_out (50000 x 64 f32)
// ---------------------------------------------------------------------------
extern "C" void kernel_launch(void* const* d_in, const int* in_sizes, int n_in,
                              void* d_out, int out_size, void* d_ws, size_t ws_size,
                              hipStream_t stream) {
  const float* h   = (const float*)d_in[0];
  const float* W   = (const float*)d_in[1];
  const float* a   = (const float*)d_in[2];
  const int*   src = (const int*)d_in[3];
  const int*   dst = (const int*)d_in[4];
  float* hout = (float*)d_out;

  // Workspace layout (bytes), ~16.8 MB total
  char* ws = (char*)d_ws;
  float*    z    = (float*)(ws);                 // 50000*64*4  = 12,800,000
  float*    e    = (float*)(ws + 12800000);      // 800000*4    =  3,200,000
  float*    el   = (float*)(ws + 16000000);      // 50000*4
  float*    er   = (float*)(ws + 16200000);      // 50000*4
  unsigned* menc = (unsigned*)(ws + 16400000);   // 50000*4
  float*    s    = (float*)(ws + 16600000);      // 50000*4

  const int edge_blk = (N_EDGES + 255) / 256;

  gat_gemm_wmma<<<(M_TILES + 1) / 2, 256, 0, stream>>>(h, W, z);
  gat_init<<<(N_NODES * OUT_DIM + 255) / 256, 256, 0, stream>>>(hout, menc, s);
  gat_scores<<<(N_NODES + 255) / 256, 256, 0, stream>>>(z, a, el, er);
  gat_edge_max<<<edge_blk, 256, 0, stream>>>(src, dst, el, er, e, menc);
  gat_edge_exp<<<edge_blk, 256, 0, stream>>>(dst, menc, e, s);
  gat_scatter<<<(N_EDGES * 32 + 255) / 256, 256, 0, stream>>>(src, dst, e, s, z, hout);
}